// PostprocessLayer_6511170421772
// MI455X (gfx1250) — compile-verified
//
#include <hip/hip_runtime.h>
#include <hip/hip_bf16.h>

// ---------------------------------------------------------------------------
// YOLO postprocess for MI455X (gfx1250, wave32).
//  - decode:  affine box decode expressed as its true GEMM form and run on
//             V_WMMA_F32_16X16X4_F32 (A = 16 boxes x 4 raw coords, B = const
//             4x4 weight padded to 4x16, C = grid-offset bias accumulator)
//  - sort:    per-batch bitonic argsort in LDS; score slice staged by the
//             CDNA5 Tensor Data Mover (tensor_load_to_lds / TENSORcnt)
//  - NMS:     per-batch greedy scan; sorted boxes gathered into LDS via the
//             CDNA5 async global->LDS path (ASYNCcnt / s_wait_asynccnt)
// ---------------------------------------------------------------------------

#define GRID_DIM   26
#define N_ANCH     5
#define BATCH      16
#define NBOX       (GRID_DIM * GRID_DIM * N_ANCH)   // 3380
#define NPAD       4096
#define IOU_T      0.4f
#define NTILE      ((NBOX + 15) / 16)               // 212 wave-tiles per batch
#define TILES_ALL  (BATCH * NTILE)                  // 3392 (divisible by 8)

typedef unsigned int u32;
typedef __attribute__((ext_vector_type(4))) u32   u32x4;
typedef __attribute__((ext_vector_type(8))) u32   u32x8;
typedef __attribute__((ext_vector_type(2))) float v2f;
typedef __attribute__((ext_vector_type(8))) float v8f;

// ---------------- Kernel 1: WMMA box decode + confidence extraction ---------
// D(16x4) = A(16x4 raw t) x W(4x4) + bias(gx,gy), one 16-box tile per wave.
__global__ __launch_bounds__(256)
void decode_wmma_kernel(const float* __restrict__ y_pred,
                        float* __restrict__ out_coord,  // [B*NBOX*4]
                        float* __restrict__ scores)     // [B*NBOX]
{
    const int lane = threadIdx.x & 31;
    const int wid  = threadIdx.x >> 5;
    const int tile = blockIdx.x * 8 + wid;      // < TILES_ALL by construction
    const int b    = tile / NTILE;
    const int row0 = (tile - b * NTILE) * 16;   // first box row of this tile

    const int half = lane >> 4;                 // 0 -> K={0,1}, 1 -> K={2,3}
    const int l15  = lane & 15;

    // ---- A: 16x4 f32; lane holds row M=l15, K-pair per lane half ----------
    int arow = row0 + l15;
    int arc  = arow < NBOX ? arow : NBOX - 1;   // clamp tail, keep EXEC full
    const float* ap = y_pred + ((size_t)b * NBOX + arc) * 25 + half * 2;
    v2f A; A.x = ap[0]; A.y = ap[1];

    // ---- B: 4x16 f32; constant decode weights, columns >= 4 are zero ------
    const float Wm[4][4] = {
        { 16.f,   0.f, 16.f,   0.f },   // t0 -> x1,x2
        {  0.f, -16.f,  0.f, -16.f },   // t1 -> y1,y2
        { -8.f,   0.f,  8.f,   0.f },   // t2 -> x1,x2 (half-width)
        {  0.f,  -8.f,  0.f,   8.f }};  // t3 -> y1,y2 (half-height)
    v2f Bv;
    Bv.x = (l15 < 4) ? Wm[half * 2 + 0][l15] : 0.f;
    Bv.y = (l15 < 4) ? Wm[half * 2 + 1][l15] : 0.f;

    // ---- C: bias accumulator; VGPR r -> row m = r + half*8, col n = l15 ---
    v8f C;
#pragma unroll
    for (int r = 0; r < 8; ++r) {
        int m    = r + half * 8;
        int row  = row0 + m;
        int rc   = row < NBOX ? row : NBOX - 1;
        int cell = rc / N_ANCH;
        int gy   = cell / GRID_DIM;
        int gx   = cell - gy * GRID_DIM;
        float bias = 0.f;
        if (l15 == 0 || l15 == 2) bias = 16.f * (float)gx;
        else if (l15 == 1 || l15 == 3) bias = 416.f - 16.f * (float)gy;
        C[r] = bias;
    }

    // EXEC is all-ones here (ISA 7.12 requirement).
    v8f D = __builtin_amdgcn_wmma_f32_16x16x4_f32(
        /*neg_a=*/false, A, /*neg_b=*/false, Bv,
        /*c_mod=*/(short)0, C, /*reuse_a=*/false, /*reuse_b=*/false);

    // ---- store D columns 0..3 = [x1,y1,x2,y2] -----------------------------
    if (l15 < 4) {
#pragma unroll
        for (int r = 0; r < 8; ++r) {
            int row = row0 + r + half * 8;
            if (row < NBOX)
                out_coord[(((size_t)b * NBOX + row) << 2) + l15] = D[r];
        }
    }

    // ---- confidence: lanes 0..15 (half 0) cover the tile's 16 rows --------
    if (half == 0) {
        int row = row0 + l15;
        if (row < NBOX)
            scores[(size_t)b * NBOX + row] =
                y_pred[((size_t)b * NBOX + row) * 25 + 4];
    }
}

// ---------------- Kernel 2: per-batch descending bitonic argsort ------------
__global__ __launch_bounds__(512)
void sort_kernel(const float* __restrict__ scores,
                 int* __restrict__ order)   // [B*NBOX] sorted indices
{
    __shared__ float key[NPAD];
    __shared__ int   idx[NPAD];

    const int b   = blockIdx.x;
    const int tid = threadIdx.x;
    const int nth = 512;

    // --- TDM: DMA this batch's 3380-float score slice straight into key[] ---
    // D# per CDNA5 ISA ch.8: group0 = {count, lds_addr, global_addr, type=2},
    // group1 = {data_size=4B, tensor_dim0=tile_dim0=3380, dim1=1, stride0=3380}.
    // Issued by wave 0 only (EXEC-ignoring, wave-level DMA; one issue suffices).
    if (tid < 32) {
        unsigned long long ga = (unsigned long long)(uintptr_t)(scores + (size_t)b * NBOX);
        u32 lds = (u32)(uintptr_t)&key[0];            // low 32b = LDS byte addr
        u32x4 g0 = { 1u,                              // count = 1 (valid)
                     lds,                             // lds_addr
                     (u32)(ga & 0xFFFFFFFFu),         // global_addr[31:0]
                     ((u32)(ga >> 32) & 0x1FFFFFFu) | (2u << 30) }; // ga[56:32] | type=2
        u32x8 g1 = { 0x2u << 16,                      // data_size = 2 -> 4 bytes
                     ((u32)NBOX & 0xFFFFu) << 16,     // tensor_dim0[15:0]
                     1u << 16,                        // tensor_dim0[31:16]=0 | tensor_dim1=1
                     ((u32)NBOX & 0xFFFFu) << 16,     // tensor_dim1[31:16]=0 | tile_dim0
                     1u,                              // tile_dim1 = 1
                     (u32)NBOX,                       // tensor_dim0_stride[31:0]
                     0u, 0u };
        asm volatile("tensor_load_to_lds %0, %1" :: "s"(g0), "s"(g1) : "memory");
#if __has_builtin(__builtin_amdgcn_s_wait_tensorcnt)
        __builtin_amdgcn_s_wait_tensorcnt(0);
#else
        asm volatile("s_wait_tensorcnt 0x0" ::: "memory");
#endif
    }
    // Padding + index init (disjoint from the TDM destination range).
    for (int i = tid; i < NPAD; i += nth) {
        if (i >= NBOX) key[i] = -1.0e30f;
        idx[i] = i;
    }
    __syncthreads();

    for (int k = 2; k <= NPAD; k <<= 1) {
        for (int j = k >> 1; j > 0; j >>= 1) {
            for (int i = tid; i < NPAD; i += nth) {
                int ixj = i ^ j;
                if (ixj > i) {
                    bool desc = ((i & k) == 0);
                    float ki = key[i], kj = key[ixj];
                    bool sw = desc ? (ki < kj) : (ki > kj);
                    if (sw) {
                        key[i] = kj; key[ixj] = ki;
                        int t = idx[i]; idx[i] = idx[ixj]; idx[ixj] = t;
                    }
                }
            }
            __syncthreads();
        }
    }
    for (int i = tid; i < NBOX; i += nth) order[b * NBOX + i] = idx[i];
}

// ---------------- Kernel 3: greedy NMS, LDS-resident per batch --------------
__global__ __launch_bounds__(1024)
void nms_kernel(const float* __restrict__ coords,   // decoded [B*NBOX*4]
                const float* __restrict__ scores,   // [B*NBOX]
                const int*   __restrict__ order,    // [B*NBOX]
                float* __restrict__ out_cls)        // [B*NBOX]
{
    __shared__ float4        box[NBOX];   // sorted order, 54 KB of 320 KB/WGP
    __shared__ unsigned char sup[NBOX];   // suppression flags

    const int b   = blockIdx.x;
    const int tid = threadIdx.x;
    const int nth = 1024;

    // Gather score-sorted boxes straight into LDS with the CDNA5 async
    // global->LDS DMA path (bypasses VGPRs, tracked by ASYNCcnt).
    const float* cbase = coords + (size_t)b * NBOX * 4;
    unsigned long long saddr = (unsigned long long)(uintptr_t)cbase;
    for (int i = tid; i < NBOX; i += nth) {
        int ord = order[b * NBOX + i];
        unsigned lds_addr = (unsigned)(uintptr_t)&box[i];   // low 32b = LDS byte addr
        unsigned voff     = (unsigned)(ord * 16);
        asm volatile("global_load_async_to_lds_b128 %0, %1, %2"
                     :: "v"(lds_addr), "v"(voff), "s"(saddr)
                     : "memory");
        sup[i] = 0;
    }
#if __has_builtin(__builtin_amdgcn_s_wait_asynccnt)
    __builtin_amdgcn_s_wait_asynccnt(0);
#else
    asm volatile("s_wait_asynccnt 0x0" ::: "memory");
#endif
    __syncthreads();

    // Greedy scan: box i (sorted) is kept iff no previously-kept box overlaps
    // it with IoU > 0.4; a kept box suppresses all later overlapping boxes.
    for (int i = 0; i < NBOX; ++i) {
        if (!sup[i]) {
            float4 bi = box[i];
            float  ai = (bi.z - bi.x) * (bi.w - bi.y);
            for (int j = i + 1 + tid; j < NBOX; j += nth) {
                if (!sup[j]) {
                    float4 bj = box[j];
                    float iw = fminf(bi.z, bj.z) - fmaxf(bi.x, bj.x);
                    float ih = fminf(bi.w, bj.w) - fmaxf(bi.y, bj.y);
                    iw = fmaxf(iw, 0.0f);
                    ih = fmaxf(ih, 0.0f);
                    float inter = iw * ih;
                    float aj  = (bj.z - bj.x) * (bj.w - bj.y);
                    float uni = ai + aj - inter;
                    float iou = (inter > 0.0f) ? (inter / uni) : 0.0f;
                    if (iou > IOU_T) sup[j] = 1;
                }
            }
        }
        __syncthreads();
    }

    // Scatter conf * keep back through the sort permutation.
    for (int i = tid; i < NBOX; i += nth) {
        int ord = order[b * NBOX + i];
        float s = scores[b * NBOX + ord];
        out_cls[b * NBOX + ord] = sup[i] ? 0.0f : s;
    }
}

// ---------------------------------------------------------------------------
extern "C" void kernel_launch(void* const* d_in, const int* in_sizes, int n_in,
                              void* d_out, int out_size, void* d_ws, size_t ws_size,
                              hipStream_t stream) {
    (void)in_sizes; (void)n_in; (void)out_size; (void)ws_size;

    const float* y_pred = (const float*)d_in[0];
    float* out       = (float*)d_out;
    float* out_coord = out;                                  // [B*NBOX*4]
    float* out_cls   = out + (size_t)BATCH * NBOX * 4;       // [B*NBOX]

    float* scores = (float*)d_ws;                                               // B*NBOX f32
    int*   order  = (int*)((char*)d_ws + (size_t)BATCH * NBOX * sizeof(float)); // B*NBOX i32

    decode_wmma_kernel<<<TILES_ALL / 8, 256, 0, stream>>>(y_pred, out_coord, scores);
    sort_kernel       <<<BATCH, 512, 0, stream>>>(scores, order);
    nms_kernel        <<<BATCH, 1024, 0, stream>>>(out_coord, scores, order, out_cls);
}